// CustomNetwork_40286793236497
// MI455X (gfx1250) — compile-verified
//
#include <hip/hip_runtime.h>
#include <hip/hip_bf16.h>
#include <math.h>

// ---------------------------------------------------------------------------
// Sizes from the reference
// ---------------------------------------------------------------------------
#define WIDTH   1024
#define HALF    512
#define DEPTH   8
#define BDEPTH  10
#define BATCH   65536
#define ROWS_PER_WG 16
#define TSTRIDE 1028                  // 16B-aligned row stride in LDS floats

// LDS byte offset of the staging buffer (dynamic LDS assumed to start at 0)
#define MATS_LDS_OFF (ROWS_PER_WG * TSTRIDE * 4)
#define MATS_BYTES   (16384 * 4)

typedef __attribute__((ext_vector_type(2))) float v2f;
typedef __attribute__((ext_vector_type(4))) float v4f;
typedef __attribute__((ext_vector_type(8))) float v8f;
typedef __attribute__((ext_vector_type(4))) unsigned int u32x4;
typedef __attribute__((ext_vector_type(4))) int          i32x4;
typedef __attribute__((ext_vector_type(8))) int          i32x8;

#if defined(__has_builtin)
#if __has_builtin(__builtin_amdgcn_tensor_load_to_lds)
#define HAVE_TDM 1
#endif
#endif

// ---------------------------------------------------------------------------
// TDM: 1-D DMA of `bytes` (multiple of 4) from global to LDS.
// D# per CDNA5 ISA ch.8: group0 = {count/type, lds_addr, global_addr},
// group1 = {wg_mask=0, data_size=4B, tensor_dim0=N, tensor_dim1=1,
//           tile_dim0=N, tile_dim1=1, dim0_stride=N, dim1_stride=N}.
// ---------------------------------------------------------------------------
#ifdef HAVE_TDM
__device__ __forceinline__ void tdm_load_1d(unsigned lds_off, const void* src,
                                            unsigned bytes) {
  const unsigned n = bytes >> 2;                 // fp32 elements
  const unsigned long long ga = (unsigned long long)src;
  u32x4 g0;
  g0.x = 1u;                                     // count=1, user mode
  g0.y = lds_off;                                // lds_addr (bytes)
  g0.z = (unsigned)(ga & 0xffffffffu);           // global_addr[31:0]
  g0.w = (unsigned)((ga >> 32) & 0x01ffffffu) |  // global_addr[56:32]
         (2u << 30);                             // type = 2 ("image")
  i32x8 g1;
  g1[0] = (int)(2u << 16);                       // data_size=2 (4B), wg_mask=0
  g1[1] = (int)((n & 0xffffu) << 16);            // tensor_dim0[15:0]
  g1[2] = (int)(((n >> 16) & 0xffffu) | (1u << 16)); // dim0[31:16], tensor_dim1=1
  g1[3] = (int)((n & 0xffffu) << 16);            // tile_dim0 = n (fits 16 bits)
  g1[4] = 1;                                     // tile_dim1 = 1
  g1[5] = (int)n;                                // tensor_dim0_stride[31:0]
  g1[6] = (int)((n & 0xffffu) << 16);            // stride[47:32]=0, dim1_stride lo
  g1[7] = 0;
  i32x4 z4 = {};
#if __clang_major__ >= 23
  i32x8 z8 = {};
  __builtin_amdgcn_tensor_load_to_lds(g0, g1, z4, z4, z8, 0);
#else
  __builtin_amdgcn_tensor_load_to_lds(g0, g1, z4, z4, 0);
#endif
}
#endif

// ---------------------------------------------------------------------------
// Kernel 0: compose per-layer block matrices from the butterfly angles.
//   Pass 0 (stages 0-3): M1[layer][g<64][16][16]
//   Pass 1 (stages 4-7): M2[layer][g<64][16][16]
//   Pass 2 (stages 8-9): M3[layer][G<64][16][16]  (block-diag of four 4x4)
// bparams flat layout: [DEPTH][HALF][BDEPTH].
// ---------------------------------------------------------------------------
__global__ void compose_mats(const float* __restrict__ bp,
                             float* __restrict__ M1,
                             float* __restrict__ M2,
                             float* __restrict__ M3) {
  int t = blockIdx.x * blockDim.x + threadIdx.x;
  if (t < 1024) {
    // 16x16 passes (stages 0-3 and 4-7)
    int pass  = t >> 9;
    int layer = (t & 511) >> 6;
    int g     = t & 63;
    int sbase = pass * 4;
    float S[16][16];
    for (int p = 0; p < 16; p++)
      for (int q = 0; q < 16; q++) S[p][q] = (p == q) ? 1.f : 0.f;
    for (int s = 0; s < 4; s++) {
      float NS[16][16];
      for (int p0 = 0; p0 < 8; p0++) {
        int p1 = p0 + 8;
        int i  = (((p0 & 7) >> s) << (6 + s)) | (g << s) | (p0 & ((1 << s) - 1));
        float th = bp[layer * (HALF * BDEPTH) + i * BDEPTH + (sbase + s)];
        float c = cosf(th), sn = sinf(th);
        int q0 = (p0 & 7) << 1;
        for (int col = 0; col < 16; col++) {
          NS[q0][col]     =  c  * S[p0][col] + sn * S[p1][col];
          NS[q0 + 1][col] = -sn * S[p0][col] + c  * S[p1][col];
        }
      }
      for (int p = 0; p < 16; p++)
        for (int col = 0; col < 16; col++) S[p][col] = NS[p][col];
    }
    float* Mo = (pass == 0 ? M1 : M2) + (layer * 64 + g) * 256;
    for (int u = 0; u < 16; u++)
      for (int c2 = 0; c2 < 16; c2++) Mo[u * 16 + c2] = S[u][c2];
  } else if (t < 1536) {
    // stages 8-9: four 4x4 rotations packed block-diagonally into 16x16
    int tt    = t - 1024;
    int layer = tt >> 6;
    int G     = tt & 63;
    float Mo[16][16];
    for (int u = 0; u < 16; u++)
      for (int v = 0; v < 16; v++) Mo[u][v] = 0.f;
    for (int q = 0; q < 4; q++) {
      int g = G * 4 + q;
      float S[4][4];
      for (int p = 0; p < 4; p++)
        for (int c2 = 0; c2 < 4; c2++) S[p][c2] = (p == c2) ? 1.f : 0.f;
      for (int s = 0; s < 2; s++) {
        float NS[4][4];
        for (int p0 = 0; p0 < 2; p0++) {
          int p1 = p0 + 2;
          int i  = (((p0 & 1) >> s) << (8 + s)) | (g << s) | (p0 & ((1 << s) - 1));
          float th = bp[layer * (HALF * BDEPTH) + i * BDEPTH + (8 + s)];
          float c = cosf(th), sn = sinf(th);
          int q0 = (p0 & 1) << 1;
          for (int col = 0; col < 4; col++) {
            NS[q0][col]     =  c  * S[p0][col] + sn * S[p1][col];
            NS[q0 + 1][col] = -sn * S[p0][col] + c  * S[p1][col];
          }
        }
        for (int p = 0; p < 4; p++)
          for (int col = 0; col < 4; col++) S[p][col] = NS[p][col];
      }
      // Mbig[q*4+l][t2*4+q] = S[l][t2]
      for (int l = 0; l < 4; l++)
        for (int t2 = 0; t2 < 4; t2++) Mo[q * 4 + l][t2 * 4 + q] = S[l][t2];
    }
    float* dst = M3 + (layer * 64 + G) * 256;
    for (int u = 0; u < 16; u++)
      for (int v = 0; v < 16; v++) dst[u * 16 + v] = Mo[u][v];
  }
}

// ---------------------------------------------------------------------------
// Kernel 1: main network. One 16-row tile per workgroup, resident in LDS.
// Per layer: three WMMA passes (pass 2 uses block-diag 16x16 with a
// different gather), then elementwise quintic activation (layers 0-6).
// ---------------------------------------------------------------------------
__global__ void __launch_bounds__(256)
butterfly_main(const float* __restrict__ X,
               const float* __restrict__ bias,
               const float* __restrict__ slope,
               const float* __restrict__ scale,
               const float* __restrict__ M1,
               const float* __restrict__ M2,
               const float* __restrict__ M3,
               float* __restrict__ out) {
  extern __shared__ float lds[];
  float* tile  = lds;                         // 16 * TSTRIDE
  float* mats  = lds + ROWS_PER_WG * TSTRIDE; // 16384 floats
  float* actb  = mats + 16384;                // 1024
  float* actsl = actb + 1024;                 // 1024
  float* actsc = actsl + 1024;                // 1024
  float* actrc = actsc + 1024;                // 1024 (1/scale)

  const int tid  = threadIdx.x;
  const int lane = tid & 31;
  const int wv   = tid >> 5;                  // wave 0..7
  const int r16  = lane & 15;
  const int hi   = lane >> 4;                 // lane half
  const long rowBase = (long)blockIdx.x * ROWS_PER_WG;

  // ---- load tile (coalesced float4) ----
  const v4f* Xv = (const v4f*)X;
  #pragma unroll
  for (int k = 0; k < 16; k++) {
    int idx = k * 256 + tid;
    int rr  = idx >> 8;
    int c4  = idx & 255;
    v4f v = Xv[(rowBase + rr) * 256 + c4];
    *(v4f*)&tile[rr * TSTRIDE + c4 * 4] = v;
  }
  __syncthreads();

  for (int layer = 0; layer < DEPTH; layer++) {
    for (int pass = 0; pass < 3; pass++) {
      const float* Msrc =
          (pass == 0 ? M1 : (pass == 1 ? M2 : M3)) + layer * 16384;

      // ---- stage 64 16x16 matrices into LDS ----
#ifdef HAVE_TDM
      if (wv == 0) {
        tdm_load_1d(MATS_LDS_OFF, Msrc, MATS_BYTES);
        __builtin_amdgcn_s_wait_tensorcnt(0);
      }
#else
      for (int k = 0; k < 64; k++) {
        int idx = k * 256 + tid;
        mats[idx] = Msrc[idx];
      }
#endif
      // stage activation params once per layer (overlaps with TDM)
      if (pass == 0 && layer < DEPTH - 1) {
        for (int k = 0; k < 4; k++) {
          int idx = k * 256 + tid;
          float b = bias [layer * WIDTH + idx];
          float s = slope[layer * WIDTH + idx];
          float c = scale[layer * WIDTH + idx];
          actb[idx]  = b;
          actsl[idx] = s;
          actsc[idx] = c;
          actrc[idx] = 1.0f / c;
        }
      }
      __syncthreads();

      // ---- WMMA: 8 groups per wave, D kept in regs for in-place update ----
      v8f acc[8];
      #pragma unroll
      for (int gi = 0; gi < 8; gi++) {
        const int g = wv * 8 + gi;
        const float* Mg = mats + g * 256;
        v8f c = {};
        #pragma unroll
        for (int k = 0; k < 4; k++) {
          const int ta = 4 * k + 2 * hi;      // K rows for this lane half
          v2f a, b;
          a.x = Mg[r16 * 16 + ta];
          a.y = Mg[r16 * 16 + ta + 1];
          int c0, c1;
          if (pass < 2) {                     // gather: col = t*64 + g
            c0 = ta * 64 + g;
            c1 = c0 + 64;
          } else {                            // gather: col = t*256 + g*4 + q
            c0 = k * 256 + g * 4 + 2 * hi;
            c1 = c0 + 1;
          }
          b.x = tile[r16 * TSTRIDE + c0];
          b.y = tile[r16 * TSTRIDE + c1];
          c = __builtin_amdgcn_wmma_f32_16x16x4_f32(
                false, a, false, b, (short)0, c, false, false);
        }
        acc[gi] = c;
      }
      __syncthreads();                        // all reads done -> in-place write
      #pragma unroll
      for (int gi = 0; gi < 8; gi++) {
        const int g = wv * 8 + gi;
        #pragma unroll
        for (int j = 0; j < 8; j++) {
          const int u = j + 8 * hi;           // D: VGPR j -> rows j / j+8
          tile[r16 * TSTRIDE + g * 16 + u] = acc[gi][j];
        }
      }
      __syncthreads();
    }

    // ---- elementwise quintic activation (in place, rcp instead of div) ----
    if (layer < DEPTH - 1) {
      #pragma unroll
      for (int k = 0; k < 16; k++) {
        int idx = k * 256 + tid;
        int rr  = idx >> 8;
        int c4  = idx & 255;
        float* p = &tile[rr * TSTRIDE + c4 * 4];
        v4f v = *(v4f*)p;
        #pragma unroll
        for (int e = 0; e < 4; e++) {
          const int col = c4 * 4 + e;
          const float sl = actsl[col], bi = actb[col];
          const float sc = actsc[col], rc = actrc[col];
          float u = (v[e] * sl - bi) * rc;
          u = fminf(1.f, fmaxf(-1.f, u));
          const float u2 = u * u;
          v[e] = sc * (u * (1.875f + u2 * (-1.25f + 0.375f * u2)));
        }
        *(v4f*)p = v;
      }
      __syncthreads();
    }
  }

  // ---- store tile (coalesced float4) ----
  v4f* Ov = (v4f*)out;
  #pragma unroll
  for (int k = 0; k < 16; k++) {
    int idx = k * 256 + tid;
    int rr  = idx >> 8;
    int c4  = idx & 255;
    Ov[(rowBase + rr) * 256 + c4] = *(v4f*)&tile[rr * TSTRIDE + c4 * 4];
  }
}

// ---------------------------------------------------------------------------
// Launcher
// ---------------------------------------------------------------------------
extern "C" void kernel_launch(void* const* d_in, const int* in_sizes, int n_in,
                              void* d_out, int out_size, void* d_ws, size_t ws_size,
                              hipStream_t stream) {
  (void)in_sizes; (void)n_in; (void)out_size; (void)ws_size;
  const float* X     = (const float*)d_in[0];
  const float* bp    = (const float*)d_in[1];
  const float* bias  = (const float*)d_in[2];
  const float* slope = (const float*)d_in[3];
  const float* scale = (const float*)d_in[4];
  float* out = (float*)d_out;

  float* M1 = (float*)d_ws;                  // 8*64*256 floats
  float* M2 = M1 + 8 * 64 * 256;
  float* M3 = M2 + 8 * 64 * 256;

  compose_mats<<<12, 128, 0, stream>>>(bp, M1, M2, M3);

  const size_t shmem =
      (ROWS_PER_WG * TSTRIDE + 16384 + 4 * 1024) * sizeof(float); // ~147.7 KB
  (void)hipFuncSetAttribute((const void*)butterfly_main,
                            hipFuncAttributeMaxDynamicSharedMemorySize,
                            (int)shmem);
  butterfly_main<<<BATCH / ROWS_PER_WG, 256, shmem, stream>>>(
      X, bias, slope, scale, M1, M2, M3, out);
}